// DFLLoss_10127532883990
// MI455X (gfx1250) — compile-verified
//
#include <hip/hip_runtime.h>

// ---------------- problem constants (from reference) ----------------
#define BINS     8
#define NCOORD   4
#define BB       32
#define HH       160
#define WW       160
#define HW       (HH * WW)            // 25600
#define CC       (NCOORD * BINS)      // 32 channels
#define TILE_PIX 256                  // pixels per TDM tile
#define NTILES   ((BB * HW) / TILE_PIX) // 3200
#define NBLK     800                  // persistent blocks
#define TPB      (NTILES / NBLK)      // 4 tiles per block
#define THREADS  256                  // 8 waves (wave32)

typedef unsigned int uint32x4 __attribute__((ext_vector_type(4)));
typedef int          int32x8  __attribute__((ext_vector_type(8)));
typedef int          int32x4  __attribute__((ext_vector_type(4)));

// Issue a TDM load of a 2D tile: 32 rows (channels) x TILE_PIX cols (pixels),
// row stride = HW floats, 4-byte elements, into LDS at lds_byte_off.
// D# layout per CDNA5 ISA ch.8 (group0: control/lds/global addr, group1: dims).
__device__ __forceinline__ void tdm_issue_tile(const float* gsrc,
                                               unsigned lds_byte_off) {
  unsigned long long ga = (unsigned long long)(size_t)gsrc;
  uint32x4 g0;
  g0.x = 1u;                                        // count=1, user D#, no gather
  g0.y = lds_byte_off;                              // lds_addr (bytes)
  g0.z = (unsigned)(ga & 0xFFFFFFFFull);            // global_addr[31:0]
  g0.w = (unsigned)((ga >> 32) & 0x01FFFFFFull)     // global_addr[56:32]
         | (2u << 30);                              // type=2 ("image")
  int32x8 g1;
  g1[0] = (int)(2u << 16);                          // wg_mask=0, data_size=2 (4B)
  g1[1] = (int)((unsigned)(TILE_PIX & 0xFFFF) << 16); // tensor_dim0 lo16 @ [63:48]
  g1[2] = (int)(((unsigned)TILE_PIX >> 16)          // tensor_dim0 hi16 @ [79:64]
                | ((unsigned)CC << 16));            // tensor_dim1 lo16 @ [95:80]
  g1[3] = (int)(((unsigned)CC >> 16)                // tensor_dim1 hi16 @ [111:96]
                | ((unsigned)TILE_PIX << 16));      // tile_dim0 @ [127:112]
  g1[4] = (int)(CC & 0xFFFF);                       // tile_dim1=32, tile_dim2=0
  g1[5] = (int)HW;                                  // tensor_dim0_stride[31:0]
  g1[6] = 0;                                        // stride0 hi, stride1 lo
  g1[7] = 0;                                        // stride1 hi
  int32x4 zz4 = {0, 0, 0, 0};                       // groups 2/3 unused (2D tile)
  int32x8 zz8 = {0, 0, 0, 0, 0, 0, 0, 0};          // extra group (clang-23 form)
  __builtin_amdgcn_tensor_load_to_lds(g0, g1, zz4, zz4, zz8, 0);
}

// Stage 1: per-block partial sums of (masked per-coord NLL, npos).
__global__ __launch_bounds__(THREADS) void dfl_partial_kernel(
    const float* __restrict__ logits,    // [B, 32, H, W]
    const float* __restrict__ targets,   // [B, HW, 4]
    const unsigned char* __restrict__ mask, // [B, HW] bool (1 byte)
    float* __restrict__ partials) {      // [2*NBLK]
  __shared__ float tile[2][CC * TILE_PIX];   // 2 x 32KB double buffer
  const int tid = threadIdx.x;
  const int wid = tid >> 5;
  const int blk = blockIdx.x;
  const int t0  = blk * TPB;

  float accL = 0.0f, accN = 0.0f;

  // Prologue: wave 0 kicks off the first tile DMA.
  if (wid == 0) {
    int p0 = t0 * TILE_PIX;
    int b = p0 / HW, hw0 = p0 % HW;
    tdm_issue_tile(logits + (size_t)b * CC * HW + hw0,
                   (unsigned)(size_t)&tile[0][0]);
  }

  for (int i = 0; i < TPB; ++i) {
    if (wid == 0) __builtin_amdgcn_s_wait_tensorcnt(0); // tile i landed in LDS
    __syncthreads();                                    // visible to all waves

    // Prefetch next tile into the other buffer (safe: that buffer's readers
    // all passed the barrier at the end of iteration i-1).
    if (wid == 0 && (i + 1) < TPB) {
      int p1 = (t0 + i + 1) * TILE_PIX;
      int b1 = p1 / HW, hw1 = p1 % HW;
      tdm_issue_tile(logits + (size_t)b1 * CC * HW + hw1,
                     (unsigned)(size_t)&tile[(i + 1) & 1][0]);
    }

    // ---- compute tile i ----
    const float* buf = &tile[i & 1][0];
    int p  = (t0 + i) * TILE_PIX;
    int b  = p / HW;
    int hw = (p % HW) + tid;
    size_t pix = (size_t)b * HW + hw;

    float4 tv = ((const float4*)targets)[pix];   // 16B coalesced
    float  mv = mask[pix] ? 1.0f : 0.0f;
    float  tk4[4] = {tv.x, tv.y, tv.z, tv.w};

    float per = 0.0f;
#pragma unroll
    for (int k = 0; k < NCOORD; ++k) {
      const float* xb = buf + (k * BINS) * TILE_PIX + tid; // bank-conflict free
      float m = xb[0];
#pragma unroll
      for (int c = 1; c < BINS; ++c) m = fmaxf(m, xb[c * TILE_PIX]);
      float s = 0.0f;
#pragma unroll
      for (int c = 0; c < BINS; ++c) s += __expf(xb[c * TILE_PIX] - m);
      float lse = m + __logf(s);                 // log-sum-exp

      float t  = fminf(fmaxf(tk4[k], 0.0f), (float)(BINS - 1) - 1e-4f);
      float fl = floorf(t);
      int   li = (int)fl;
      int   ui = (li + 1 < BINS - 1) ? (li + 1) : (BINS - 1);
      float wu = t - fl, wl = 1.0f - wu;
      // nll(bin) = lse - logit(bin); gather the two taps from LDS
      per += wl * (lse - xb[li * TILE_PIX]) + wu * (lse - xb[ui * TILE_PIX]);
    }
    accL += per * mv;
    accN += mv;
    __syncthreads();   // done reading buf[i&1] before it is overwritten
  }

  // ---- deterministic block reduction ----
  for (int off = 16; off > 0; off >>= 1) {
    accL += __shfl_down(accL, off, 32);
    accN += __shfl_down(accN, off, 32);
  }
  float* red = &tile[0][0];                       // reuse LDS (post-barrier)
  if ((tid & 31) == 0) { red[wid] = accL; red[8 + wid] = accN; }
  __syncthreads();
  if (tid == 0) {
    float sL = 0.0f, sN = 0.0f;
#pragma unroll
    for (int w = 0; w < THREADS / 32; ++w) { sL += red[w]; sN += red[8 + w]; }
    partials[blk]        = sL;
    partials[NBLK + blk] = sN;
  }
}

// Stage 2: fixed-order reduction of the 800 partial pairs -> scalar loss.
__global__ __launch_bounds__(256) void dfl_finalize_kernel(
    const float* __restrict__ partials, float* __restrict__ out) {
  __shared__ float rL[256], rN[256];
  const int tid = threadIdx.x;
  float sL = 0.0f, sN = 0.0f;
  for (int i = tid; i < NBLK; i += 256) {
    sL += partials[i];
    sN += partials[NBLK + i];
  }
  rL[tid] = sL; rN[tid] = sN;
  __syncthreads();
  for (int off = 128; off > 0; off >>= 1) {
    if (tid < off) { rL[tid] += rL[tid + off]; rN[tid] += rN[tid + off]; }
    __syncthreads();
  }
  if (tid == 0) {
    float n    = rN[0];
    float loss = rL[0] / (fmaxf(n, 1.0f) * 4.0f);
    out[0] = (n > 0.0f) ? loss : 0.0f;
  }
}

extern "C" void kernel_launch(void* const* d_in, const int* in_sizes, int n_in,
                              void* d_out, int out_size, void* d_ws, size_t ws_size,
                              hipStream_t stream) {
  (void)in_sizes; (void)n_in; (void)out_size; (void)ws_size;
  const float*         logits  = (const float*)d_in[0];
  const float*         targets = (const float*)d_in[1];
  const unsigned char* mask    = (const unsigned char*)d_in[2];
  float*               parts   = (float*)d_ws;   // 2*NBLK floats = 6.4 KB

  dfl_partial_kernel<<<NBLK, THREADS, 0, stream>>>(logits, targets, mask, parts);
  dfl_finalize_kernel<<<1, 256, 0, stream>>>(parts, (float*)d_out);
}